// SuperpixelPooling_6880537608986
// MI455X (gfx1250) — compile-verified
//
#include <hip/hip_runtime.h>

// ---------------------------------------------------------------------------
// Superpixel mean pooling, MI455X (gfx1250).
// x: [B=4, C=128, H=512, W=512] f32   labels: [B,1,H,W] int32
// out: [B, 512, 128] f32 mean feature per label.
//
// HBM floor: 512 MiB x-read @ 23.3 TB/s ~= 23 us -> pure scatter-reduce.
//  - whole 512x129 f32 accumulator (258 KiB) lives in LDS (CDNA5 320KB/WGP,
//    impossible on CDNA4's 64KB/CU)
//  - labels slice prefetched with global_load_async_to_lds_b128 (ASYNCcnt),
//    overlapped with accumulator zeroing, s_wait_asynccnt before use
//  - x read as nontemporal float4 (lanes = 4 consecutive pixels each,
//    512B/wave-instruction, full cacheline use, no L2 pollution)
//  - ds_add_f32 scatter, row stride 129 (odd) -> bank = (label + c) % 64,
//    spread across banks for random labels at fixed c
// ---------------------------------------------------------------------------

typedef float  vf4 __attribute__((ext_vector_type(4)));   // clang ext-vector
typedef int    vi4 __attribute__((ext_vector_type(4)));

#define Bv 4
#define Cv 128
#define Hv 512
#define Wv 512
#define Pv (Hv * Wv)                 // 262144 pixels per image
#define Lv 512                       // labels
#define SLICES 64                    // workgroups per image
#define PX_PER_WG (Pv / SLICES)      // 4096
#define THREADS 256                  // 8 waves (wave32)
#define NWAVES (THREADS / 32)
#define PX_PER_WAVE (PX_PER_WG / NWAVES)   // 512
#define PX_PER_BLK (32 * 4)                // 128 px per wave iteration
#define BLKS_PER_WAVE (PX_PER_WAVE / PX_PER_BLK)  // 4
#define STRIDE (Cv + 1)              // 129: col 128 holds the count
#define ACC_ELEMS (Lv * STRIDE)      // 66048 f32 = 258 KiB
#define ACC_VEC4 (ACC_ELEMS / 4)     // 16512 float4
#define LAB_BYTES (PX_PER_WG * 4)    // 16 KiB int32 labels
#define LAB_CHUNKS (LAB_BYTES / 16)  // b128 chunks = 1024
#define CHUNKS_PER_THREAD (LAB_CHUNKS / THREADS)  // 4

__global__ __launch_bounds__(THREADS, 1)
void sp_pool_kernel(const float* __restrict__ x, const int* __restrict__ lab,
                    float* __restrict__ sums, float* __restrict__ cnt)
{
    __shared__ float s_acc[ACC_ELEMS];     // 258 KiB accumulator
    __shared__ int   s_lab[PX_PER_WG];     // 16 KiB labels (async-filled)

    const int wg = blockIdx.x;
    const int b  = wg / SLICES;
    const int sl = wg - b * SLICES;
    const int p0 = sl * PX_PER_WG;
    const int t  = threadIdx.x;

    // ---- 1) kick off async global->LDS prefetch of this slice's labels ----
    {
        // generic shared-pointer low 32 bits == LDS byte offset (aperture layout)
        unsigned lds_base = (unsigned)(unsigned long long)(&s_lab[0]);
        unsigned long long gbase =
            (unsigned long long)(lab + (size_t)b * Pv + p0);
        #pragma unroll
        for (int i = 0; i < CHUNKS_PER_THREAD; ++i) {
            unsigned off   = (unsigned)((t + i * THREADS) * 16);
            unsigned laddr = lds_base + off;
            asm volatile("global_load_async_to_lds_b128 %0, %1, %2"
                         :: "v"(laddr), "v"(off), "s"(gbase)
                         : "memory");
        }
    }

    // ---- 2) zero the accumulator while the async engine streams labels ----
    {
        vf4* accv = (vf4*)s_acc;
        const vf4 z = {0.f, 0.f, 0.f, 0.f};
        for (int i = t; i < ACC_VEC4; i += THREADS) accv[i] = z;
    }

    asm volatile("s_wait_asynccnt 0x0" ::: "memory");
    __syncthreads();

    // ---- 3) scatter-accumulate: lane owns 4 consecutive pixels ----
    const int wave = t >> 5;
    const int lane = t & 31;
    const float* xb = x + (size_t)b * Cv * Pv + p0;   // channel 0 of slice

    for (int blk = 0; blk < BLKS_PER_WAVE; ++blk) {
        const int pi0 = wave * PX_PER_WAVE + blk * PX_PER_BLK + lane * 4;
        const vi4 lv = *(const vi4*)&s_lab[pi0];       // ds_load_b128
        float* r0 = &s_acc[lv.x * STRIDE];
        float* r1 = &s_acc[lv.y * STRIDE];
        float* r2 = &s_acc[lv.z * STRIDE];
        float* r3 = &s_acc[lv.w * STRIDE];
        atomicAdd(&r0[Cv], 1.0f);                      // count column
        atomicAdd(&r1[Cv], 1.0f);
        atomicAdd(&r2[Cv], 1.0f);
        atomicAdd(&r3[Cv], 1.0f);
        const float* xp = xb + pi0;
        #pragma unroll 4
        for (int c = 0; c < Cv; ++c) {
            // 512B/wave nontemporal streaming load of x
            const vf4 v =
                __builtin_nontemporal_load((const vf4*)(xp + (size_t)c * Pv));
            atomicAdd(&r0[c], v.x);                    // ds_add_f32 x4
            atomicAdd(&r1[c], v.y);
            atomicAdd(&r2[c], v.z);
            atomicAdd(&r3[c], v.w);
        }
    }
    __syncthreads();

    // ---- 4) flush LDS accumulator to global with float atomics ----
    float* sums_b = sums + (size_t)b * Lv * Cv;
    float* cnt_b  = cnt + b * Lv;
    for (int i = t; i < ACC_ELEMS; i += THREADS) {
        int l = i / STRIDE;
        int k = i - l * STRIDE;
        float v = s_acc[i];
        if (k == Cv) unsafeAtomicAdd(&cnt_b[l], v);
        else         unsafeAtomicAdd(&sums_b[l * Cv + k], v);
    }
}

__global__ void sp_zero_kernel(float* __restrict__ sums, float* __restrict__ cnt)
{
    int i = blockIdx.x * blockDim.x + threadIdx.x;
    if (i < Bv * Lv * Cv) sums[i] = 0.0f;
    if (i < Bv * Lv)      cnt[i]  = 0.0f;
}

__global__ void sp_div_kernel(float* __restrict__ out, const float* __restrict__ cnt)
{
    int i = blockIdx.x * blockDim.x + threadIdx.x;
    if (i >= Bv * Lv * Cv) return;
    int bl = i / Cv;                       // Cv = 128 -> shift
    out[i] *= __frcp_rn(fmaxf(cnt[bl], 1.0f));
}

extern "C" void kernel_launch(void* const* d_in, const int* in_sizes, int n_in,
                              void* d_out, int out_size, void* d_ws, size_t ws_size,
                              hipStream_t stream)
{
    const float* x      = (const float*)d_in[0];
    const int*   labels = (const int*)d_in[1];
    float*       out    = (float*)d_out;       // [B, L, C] sums -> means
    float*       cnt    = (float*)d_ws;        // [B, L] counts (8 KiB)

    const int n = Bv * Lv * Cv;
    sp_zero_kernel<<<(n + 255) / 256, 256, 0, stream>>>(out, cnt);
    sp_pool_kernel<<<Bv * SLICES, THREADS, 0, stream>>>(x, labels, out, cnt);
    sp_div_kernel<<<(n + 255) / 256, 256, 0, stream>>>(out, cnt);
}